// SlicedWassersteinDistance_8804682957212
// MI455X (gfx1250) — compile-verified
//
#include <hip/hip_runtime.h>
#include <hip/hip_bf16.h>

// ---------------------------------------------------------------------------
// SlicedWassersteinDistance for MI455X (gfx1250).
// B=16, C=3, H=W=256 -> HW=65536, P=64, BC=B*C=48.
//
// Memory-bound: theta is 805 MB f32, read once (~35us at 23.3 TB/s).
// Single fused streaming pass over theta computing, per (b,c,p):
//     stt = sum theta^2   (VALU f32 fma on already-loaded regs)
//     s1  = <f1, theta>   (V_WMMA_F32_16X16X4_F32, full f32 precision)
//     s2  = <f2, theta>   (same WMMA, column 1 of B)
// B-matrix construction is branch-free (pointer select + v_cndmask zeroing)
// so the hot loop has no EXEC-mask manipulation.
// Then a tiny finalize: p = s / sqrt(stt), bitonic sort of 64 values,
// sum of squared diffs, sqrt, mean over 48.
// No atomics anywhere -> bitwise deterministic.
// ---------------------------------------------------------------------------

typedef __attribute__((ext_vector_type(2))) float v2f;
typedef __attribute__((ext_vector_type(8))) float v8f;

#define HW    65536
#define NP    64
#define NBC   48

// ---------------------------------------------------------------------------
// Kernel 1: streaming projection pass.
// grid = (KC, 48), block = 128 (4 waves). Wave w handles p-rows [16w,16w+16)
// over this block's K-chunk. Partials written to unique slots:
//   partials[(chunkIdx*48 + bc)*192 + {0..63:s1, 64..127:s2, 128..191:stt}]
// ---------------------------------------------------------------------------
__global__ __launch_bounds__(128) void swd_proj_kernel(
    const float* __restrict__ f1,     // [48, HW]
    const float* __restrict__ f2,     // [48, HW]
    const float* __restrict__ theta,  // [48, 64, HW]
    float* __restrict__ partials,
    int chunk)                        // K elements per block (multiple of 32)
{
    const int tid    = threadIdx.x;
    const int wave   = tid >> 5;
    const int lane   = tid & 31;
    const int bc     = blockIdx.y;
    const int kchunk = blockIdx.x;

    const int p_base = wave * 16;
    const int row    = p_base + (lane & 15);  // A-matrix: lane -> M row
    const int n      = lane & 15;             // B-matrix: lane -> N col
    const int koff   = (lane < 16) ? 0 : 2;   // VGPR-half -> K offset

    const size_t kbase = (size_t)kchunk * (size_t)chunk + (size_t)koff;

    // A rows: theta[bc][row][...]
    const float* __restrict__ trow =
        theta + ((size_t)bc * NP + (size_t)row) * (size_t)HW + kbase;
    // B cols: col 0 = f1, col 1 = f2, cols 2..15 = 0.
    // Pointer selected ONCE; lanes n>=2 read f1 (same cacheline as lane 0)
    // and their values are zeroed with a branch-free select (v_cndmask).
    const float* __restrict__ gb =
        ((n == 1) ? f2 : f1) + (size_t)bc * (size_t)HW + kbase;
    const bool bvalid = (n < 2);

    v8f  c  = {};     // D/C accumulator: 16x16 f32
    float tt = 0.0f;  // per-lane partial of theta self-dot

    for (int i = 0; i < chunk; i += 32) {
        #pragma unroll
        for (int s = 0; s < 8; ++s) {
            const int o = i + 4 * s;
            // A tile: 16 rows x 4 K of theta (8B per lane, global_load_b64)
            v2f a    = *(const v2f*)(trow + o);
            v2f braw = *(const v2f*)(gb + o);
            v2f b;
            b.x = bvalid ? braw.x : 0.0f;   // v_cndmask, no EXEC divergence
            b.y = bvalid ? braw.y : 0.0f;

            // theta self-dot rides along on VALU
            tt = fmaf(a.x, a.x, tt);
            tt = fmaf(a.y, a.y, tt);

            // D = A*B + C, full f32
            c = __builtin_amdgcn_wmma_f32_16x16x4_f32(
                    false, a, false, b, (short)0, c, false, false);
        }
    }

    // ---- write partials (one writer per slot, no atomics) ----
    float* __restrict__ base = partials + ((size_t)kchunk * NBC + (size_t)bc) * 192;

    // theta self-dot: row r total = lane r + lane r+16
    float tt_other = __shfl_xor(tt, 16, 32);
    if (lane < 16)
        base[128 + p_base + lane] = tt + tt_other;

    // D layout: VGPR r, lanes 0-15 -> (M=r, N=lane), lanes 16-31 -> (M=r+8, N=lane-16)
    union { v8f v; float f[8]; } u;
    u.v = c;
    if (n < 2) {  // lanes 0,1,16,17 hold the useful columns
        const int rbase = p_base + ((lane & 16) ? 8 : 0);
        float* __restrict__ dst = base + n * 64 + rbase;  // n==0 -> s1, n==1 -> s2
        #pragma unroll
        for (int r = 0; r < 8; ++r)
            dst[r] = u.f[r];
    }
}

// ---------------------------------------------------------------------------
// Kernel 2: per-(b,c) finalize. 48 blocks x 64 threads.
// Fixed-order chunk reduction -> normalize -> bitonic sort -> swd -> sqrt.
// ---------------------------------------------------------------------------
__global__ __launch_bounds__(64) void swd_finalize_kernel(
    const float* __restrict__ partials,
    float* __restrict__ per_bc,
    int kc)
{
    const int bc = blockIdx.x;
    const int p  = threadIdx.x;  // 0..63

    float s1 = 0.0f, s2 = 0.0f, stt = 0.0f;
    for (int ch = 0; ch < kc; ++ch) {  // fixed order -> deterministic
        const float* __restrict__ base = partials + ((size_t)ch * NBC + (size_t)bc) * 192;
        s1  += base[p];
        s2  += base[64 + p];
        stt += base[128 + p];
    }

    const float inv = 1.0f / sqrtf(stt);

    __shared__ float a1[NP];
    __shared__ float a2[NP];
    __shared__ float red[NP];
    a1[p] = s1 * inv;
    a2[p] = s2 * inv;
    __syncthreads();

    // bitonic sort (ascending) of both 64-vectors
    for (int k = 2; k <= NP; k <<= 1) {
        for (int j = k >> 1; j > 0; j >>= 1) {
            const int ixj = p ^ j;
            if (ixj > p) {
                const bool asc = ((p & k) == 0);
                float x = a1[p], y = a1[ixj];
                if ((x > y) == asc) { a1[p] = y; a1[ixj] = x; }
                x = a2[p]; y = a2[ixj];
                if ((x > y) == asc) { a2[p] = y; a2[ixj] = x; }
            }
            __syncthreads();
        }
    }

    const float d = a1[p] - a2[p];
    red[p] = d * d;
    __syncthreads();
    for (int s = 32; s > 0; s >>= 1) {
        if (p < s) red[p] += red[p + s];
        __syncthreads();
    }
    if (p == 0)
        per_bc[bc] = sqrtf(red[0]);
}

// ---------------------------------------------------------------------------
// Kernel 3: mean of the 48 per-(b,c) distances, fixed order.
// ---------------------------------------------------------------------------
__global__ void swd_mean_kernel(const float* __restrict__ per_bc,
                                float* __restrict__ out)
{
    if (threadIdx.x == 0 && blockIdx.x == 0) {
        float s = 0.0f;
        for (int i = 0; i < NBC; ++i)
            s += per_bc[i];
        out[0] = s / (float)NBC;
    }
}

// ---------------------------------------------------------------------------
extern "C" void kernel_launch(void* const* d_in, const int* in_sizes, int n_in,
                              void* d_out, int out_size, void* d_ws, size_t ws_size,
                              hipStream_t stream)
{
    const float* batch1 = (const float*)d_in[0];  // [16,3,256,256]
    const float* batch2 = (const float*)d_in[1];  // [16,3,256,256]
    const float* theta  = (const float*)d_in[2];  // [16,3,64,65536]
    float*       out    = (float*)d_out;          // scalar

    // Choose K-chunk count (power of two) that fits the workspace.
    // ws_size is constant across calls -> same work every call (deterministic).
    int kc = 64;
    while (kc > 1 && ((size_t)kc * NBC * 192 + NBC) * sizeof(float) > ws_size)
        kc >>= 1;
    const int chunk = HW / kc;

    float* partials = (float*)d_ws;                        // [kc][48][192]
    float* per_bc   = partials + (size_t)kc * NBC * 192;   // [48]

    dim3 grid1(kc, NBC);
    swd_proj_kernel<<<grid1, 128, 0, stream>>>(batch1, batch2, theta, partials, chunk);
    swd_finalize_kernel<<<NBC, 64, 0, stream>>>(partials, per_bc, kc);
    swd_mean_kernel<<<1, 64, 0, stream>>>(per_bc, out);
}